// GaussianCategoricalSampler_41137196761695
// MI455X (gfx1250) — compile-verified
//
#include <hip/hip_runtime.h>

// GaussianCategoricalSampler for MI455X (gfx1250).
//
// Memory-bound: 125.8 MB in + 16.8 MB out => ~6.1 us roofline @ 23.3 TB/s.
// No matmul structure exists (per-segment argmax), so WMMA is not applicable;
// instead we exercise the CDNA5 async data-mover path:
//   global_load_async_to_lds_b128  (ASYNCcnt-tracked, no VGPR round trip)
//   s_wait_asynccnt
// followed by LDS argmax scans and coalesced b128 stores.

#define BATCH        65536
#define WIDTH        480      // input floats per row
#define OUT_W        64       // output floats per row
#define ROWS_PER_BLK 32
#define THREADS      256
#define ROW_STRIDE   484      // LDS floats per row: 484*4 = 1936 B (16B aligned,
                              // stride mod 64 banks = 36 -> <=2-way conflicts)
#define F4_PER_ROW   (WIDTH / 4)                       // 120
#define TOTAL_F4     (ROWS_PER_BLK * F4_PER_ROW)       // 3840
#define F4_ITERS     (TOTAL_F4 / THREADS)              // 15

__global__ __launch_bounds__(THREADS)
void gcs_argmax_kernel(const float* __restrict__ in, float* __restrict__ out) {
    __shared__ float smem[ROWS_PER_BLK * ROW_STRIDE];

    const int t = threadIdx.x;
    const long long blockRow = (long long)blockIdx.x * ROWS_PER_BLK;
    const float* gbase = in + blockRow * (long long)WIDTH;

    // ---- Phase 1: async DMA global -> LDS, 16B per lane, fully coalesced ----
    #pragma unroll
    for (int k = 0; k < F4_ITERS; ++k) {
        const int j      = t + k * THREADS;          // float4 index within tile
        const int row    = j / F4_PER_ROW;
        const int within = j - row * F4_PER_ROW;
        // Low 32 bits of a generic LDS pointer are the LDS byte address.
        const unsigned lds_addr =
            (unsigned)(size_t)&smem[row * ROW_STRIDE + within * 4];
        const float* gp = gbase + (size_t)j * 4;     // contiguous across j
        asm volatile("global_load_async_to_lds_b128 %0, %1, off"
                     :: "v"(lds_addr), "v"(gp)
                     : "memory");
    }
    asm volatile("s_wait_asynccnt 0" ::: "memory");  // our wave's DMAs done
    __syncthreads();                                 // whole tile visible

    // ---- Phase 2: per-segment argmax from LDS ----
    // thread t -> row (t>>3), output cols [ (t&7)*8 , +8 )  == groups 2*(t&7), +1
    const int   r   = t >> 3;
    const int   cg  = t & 7;
    const float* srow = &smem[r * ROW_STRIDE];

    float res[8];
    #pragma unroll
    for (int gi = 0; gi < 2; ++gi) {
        const int base = (cg * 2 + gi) * 30;   // 30 input cols per [1,4,8,16] group

        // size-1 segment: output mu = first of its 2 columns
        res[gi * 4 + 0] = srow[base + 0];

        // size-4 segment at base+2
        {
            float best = srow[base + 2]; int bi = 0;
            #pragma unroll
            for (int i = 1; i < 4; ++i) {
                float v = srow[base + 2 + i];
                if (v > best) { best = v; bi = i; }   // strict > == first-max
            }
            res[gi * 4 + 1] = (float)bi;
        }
        // size-8 segment at base+6
        {
            float best = srow[base + 6]; int bi = 0;
            #pragma unroll
            for (int i = 1; i < 8; ++i) {
                float v = srow[base + 6 + i];
                if (v > best) { best = v; bi = i; }
            }
            res[gi * 4 + 2] = (float)bi;
        }
        // size-16 segment at base+14
        {
            float best = srow[base + 14]; int bi = 0;
            #pragma unroll
            for (int i = 1; i < 16; ++i) {
                float v = srow[base + 14 + i];
                if (v > best) { best = v; bi = i; }
            }
            res[gi * 4 + 3] = (float)bi;
        }
    }

    // ---- Phase 3: coalesced stores ----
    // global float index = (blockRow + r)*64 + cg*8 = blockRow*64 + t*8
    float* op = out + (blockRow + r) * (long long)OUT_W + cg * 8;
    float4 o0 = make_float4(res[0], res[1], res[2], res[3]);
    float4 o1 = make_float4(res[4], res[5], res[6], res[7]);
    *(float4*)(op)     = o0;   // wave covers a contiguous 1 KB output region
    *(float4*)(op + 4) = o1;
}

extern "C" void kernel_launch(void* const* d_in, const int* in_sizes, int n_in,
                              void* d_out, int out_size, void* d_ws, size_t ws_size,
                              hipStream_t stream) {
    (void)in_sizes; (void)n_in; (void)out_size; (void)d_ws; (void)ws_size;
    const float* in  = (const float*)d_in[0];
    float*       out = (float*)d_out;
    const dim3 grid(BATCH / ROWS_PER_BLK);   // 2048 blocks
    gcs_argmax_kernel<<<grid, THREADS, 0, stream>>>(in, out);
}